// NeuralNetwork_19688130085108
// MI455X (gfx1250) — compile-verified
//
#include <hip/hip_runtime.h>
#include <math.h>

// ---------------------------------------------------------------------------
// 3-layer GAT (PyG GATConv semantics) for MI455X / gfx1250.
// Everything is L2-resident (~110 MB working set vs 192 MB L2), so kernels
// are organized around L2 atomic throughput; the dense GEMMs use the fp32
// WMMA path (v_wmma_f32_16x16x4_f32), wave32, one wave per 16x64 output tile.
// ---------------------------------------------------------------------------

typedef __attribute__((ext_vector_type(2))) float v2f;
typedef __attribute__((ext_vector_type(8))) float v8f;

#define SLOPE 0.2f               // both NEG_SLOPE and ATT_SLOPE are 0.2
#define KEY_NEG_MAX 0x00800000u  // fkey(-FLT_MAX): init value for segment max

__device__ __forceinline__ float lrelu(float x) { return x > 0.f ? x : SLOPE * x; }

// order-preserving float <-> uint keys so segment-max can use atomicMax(u32)
__device__ __forceinline__ unsigned fkey(float f) {
  unsigned u = __float_as_uint(f);
  return (u & 0x80000000u) ? ~u : (u | 0x80000000u);
}
__device__ __forceinline__ float funkey(unsigned k) {
  unsigned u = (k & 0x80000000u) ? (k ^ 0x80000000u) : ~k;
  return __uint_as_float(u);
}

__device__ __forceinline__ void edge_nodes(int e, const int* __restrict__ esrc,
                                           const int* __restrict__ edst, int E,
                                           int& s, int& d) {
  if (e < E) { s = esrc[e]; d = edst[e]; }
  else       { s = e - E;  d = e - E; }   // self loop
}

// --- per-layer scratch init: acc=0, mkey=-inf key, denom=0 ------------------
__global__ void init_layer_k(float* __restrict__ acc, unsigned* __restrict__ mkey,
                             float* __restrict__ denom, int n64, int nh) {
  int i = blockIdx.x * blockDim.x + threadIdx.x;
  if (i < n64) acc[i] = 0.f;
  if (i < nh) { mkey[i] = KEY_NEG_MAX; denom[i] = 0.f; }
}

// --- dense GEMM: out[N,64] = in[N,64] @ W[64,64] via f32 WMMA ---------------
// One wave computes a 16(row) x 64(col) tile: 4 col-tiles x 16 K-steps.
// Requires N % 16 == 0 (N = 100000 -> 6250 row tiles). Whole-wave early exit
// keeps EXEC all-1s for every wave that reaches a WMMA.
__global__ void gemm64_wmma_k(const float* __restrict__ in, const float* __restrict__ W,
                              float* __restrict__ out, int N) {
  int lane = threadIdx.x & 31;
  int wave = blockIdx.x * (blockDim.x >> 5) + (threadIdx.x >> 5);
  int row0 = wave * 16;
  if (row0 >= N) return;                  // wave-uniform branch
  int hsel = lane >> 4;                   // 0: K pair {0,1}; 1: K pair {2,3}
  int lr   = lane & 15;                   // row (A) / col (B)

  const float* arow = in + (size_t)(row0 + lr) * 64;
  v8f accd[4] = {};                       // 4 independent 16x16 accumulators

  for (int k = 0; k < 64; k += 4) {
    v2f a;
    a.x = arow[k + 2 * hsel + 0];
    a.y = arow[k + 2 * hsel + 1];
    const float* wk = W + (size_t)(k + 2 * hsel) * 64 + lr;
#pragma unroll
    for (int t = 0; t < 4; ++t) {
      v2f b;
      b.x = wk[t * 16];                   // row k+2*hsel
      b.y = wk[64 + t * 16];              // row k+2*hsel+1
      accd[t] = __builtin_amdgcn_wmma_f32_16x16x4_f32(
          /*neg_a=*/false, a, /*neg_b=*/false, b,
          /*c_mod=*/(short)0, accd[t], /*reuse_a=*/false, /*reuse_b=*/false);
    }
  }

  // D layout: VGPR v -> row (row0 + hsel*8 + v), col lr (within col tile t)
  float* orow = out + (size_t)(row0 + hsel * 8) * 64 + lr;
#pragma unroll
  for (int t = 0; t < 4; ++t)
#pragma unroll
    for (int v = 0; v < 8; ++v)
      orow[(size_t)v * 64 + t * 16] = accd[t][v];
}

// --- attention coefficients: alpha_s/d[n,h] = <xt[n,h,:], a_src/dst[h,:]> ---
__global__ void attn_coeff_k(const float* __restrict__ xt, const float* __restrict__ a_src,
                             const float* __restrict__ a_dst, float* __restrict__ als,
                             float* __restrict__ ald, int N, int H, int hshift, int C) {
  int i = blockIdx.x * blockDim.x + threadIdx.x;
  if (i >= (N << hshift)) return;
  int n = i >> hshift, h = i & (H - 1);
  const float* xr  = xt + (size_t)n * 64 + h * C;
  const float* asr = a_src + h * C;
  const float* adr = a_dst + h * C;
  float s = 0.f, d = 0.f;
  for (int c = 0; c < C; ++c) { float v = xr[c]; s += v * asr[c]; d += v * adr[c]; }
  als[i] = s;
  ald[i] = d;
}

// --- segment max of edge logits over dst --------------------------------
__global__ void edge_max_k(const int* __restrict__ esrc, const int* __restrict__ edst,
                           const float* __restrict__ als, const float* __restrict__ ald,
                           unsigned* __restrict__ mkey, int E, int E2, int H, int hshift) {
  int i = blockIdx.x * blockDim.x + threadIdx.x;
  if (i >= (E2 << hshift)) return;
  int e = i >> hshift, h = i & (H - 1);
  int s, d; edge_nodes(e, esrc, edst, E, s, d);
  float ev = lrelu(als[(s << hshift) + h] + ald[(d << hshift) + h]);
  atomicMax(&mkey[(d << hshift) + h], fkey(ev));
}

// --- p = exp(e - m[dst]); segment sum of p over dst ----------------------
__global__ void edge_sum_k(const int* __restrict__ esrc, const int* __restrict__ edst,
                           const float* __restrict__ als, const float* __restrict__ ald,
                           const unsigned* __restrict__ mkey, float* __restrict__ denom,
                           float* __restrict__ pbuf, int E, int E2, int H, int hshift) {
  int i = blockIdx.x * blockDim.x + threadIdx.x;
  if (i >= (E2 << hshift)) return;
  int e = i >> hshift, h = i & (H - 1);
  int s, d; edge_nodes(e, esrc, edst, E, s, d);
  float ev = lrelu(als[(s << hshift) + h] + ald[(d << hshift) + h]);
  float m  = funkey(mkey[(d << hshift) + h]);
  float p  = __expf(ev - m);
  pbuf[i] = p;                            // i == e*H + h
  atomicAdd(&denom[(d << hshift) + h], p);
}

// --- aggregation: acc[dst,:] += alpha * xt[src,:] ------------------------
// 16 lanes per edge, float4 per lane -> coalesced 256B gather + scatter-add.
__global__ void edge_agg_k(const int* __restrict__ esrc, const int* __restrict__ edst,
                           const float* __restrict__ xt, const float* __restrict__ pbuf,
                           const float* __restrict__ denom, float* __restrict__ acc,
                           int E, int E2, int hshift, int cshift) {
  int i = blockIdx.x * blockDim.x + threadIdx.x;
  if (i >= (E2 << 4)) return;
  int e  = i >> 4;
  int c0 = (i & 15) << 2;                 // channel group of 4
  int h  = c0 >> cshift;                  // head = c0 / C
  int s, d; edge_nodes(e, esrc, edst, E, s, d);
  float alpha = pbuf[(e << hshift) + h] / (denom[(d << hshift) + h] + 1e-16f);
  const float4 v = *(const float4*)(xt + (size_t)s * 64 + c0);
  float* ap = acc + (size_t)d * 64 + c0;
  atomicAdd(ap + 0, alpha * v.x);
  atomicAdd(ap + 1, alpha * v.y);
  atomicAdd(ap + 2, alpha * v.z);
  atomicAdd(ap + 3, alpha * v.w);
}

// --- bias + inter-layer leaky relu ---------------------------------------
__global__ void bias_act_k(const float* __restrict__ acc, const float* __restrict__ b,
                           float* __restrict__ out, int n64) {
  int i = blockIdx.x * blockDim.x + threadIdx.x;
  if (i >= n64) return;
  out[i] = lrelu(acc[i] + b[i & 63]);
}

// --- final mean over nodes -----------------------------------------------
__global__ void zero_out_k(float* __restrict__ out) {
  if (threadIdx.x < 64) out[threadIdx.x] = 0.f;
}

__global__ void mean_k(const float* __restrict__ h, float* __restrict__ out,
                       int N, float invN) {
  __shared__ float sm[256];
  int t = threadIdx.x;
  int c = t & 63;
  int g = t >> 6;                          // 4 row-groups per block
  float s = 0.f;
  for (int r = blockIdx.x * 4 + g; r < N; r += gridDim.x * 4)
    s += h[(size_t)r * 64 + c];
  sm[t] = s;
  __syncthreads();
  if (t < 64) {
    float tot = sm[t] + sm[t + 64] + sm[t + 128] + sm[t + 192];
    atomicAdd(&out[c], tot * invN);
  }
}

// ---------------------------------------------------------------------------
extern "C" void kernel_launch(void* const* d_in, const int* in_sizes, int n_in,
                              void* d_out, int out_size, void* d_ws, size_t ws_size,
                              hipStream_t stream) {
  const float* x   = (const float*)d_in[0];
  const int*   ei  = (const int*)d_in[1];
  const float* W0  = (const float*)d_in[2];
  const float* as0 = (const float*)d_in[3];
  const float* ad0 = (const float*)d_in[4];
  const float* b0  = (const float*)d_in[5];
  const float* W1  = (const float*)d_in[6];
  const float* as1 = (const float*)d_in[7];
  const float* ad1 = (const float*)d_in[8];
  const float* b1  = (const float*)d_in[9];
  const float* W2  = (const float*)d_in[10];
  const float* as2 = (const float*)d_in[11];
  const float* ad2 = (const float*)d_in[12];
  const float* b2  = (const float*)d_in[13];

  const int N  = in_sizes[0] / 64;   // 100000 (divisible by 16)
  const int E  = in_sizes[1] / 2;    // 1600000
  const int E2 = E + N;              // edges + self loops
  const int* esrc = ei;
  const int* edst = ei + E;

  // workspace layout (fp32 slots), ~110 MB total — fully L2-resident
  float* ws = (float*)d_ws;
  size_t o = 0;
  float*    hbuf  = ws + o; o += (size_t)N * 64;   // activated features (ping buffer)
  float*    xt    = ws + o; o += (size_t)N * 64;   // W-transformed features
  float*    acc   = ws + o; o += (size_t)N * 64;   // aggregation accumulator
  float*    als   = ws + o; o += (size_t)N * 4;    // alpha_src  [N,H]
  float*    ald   = ws + o; o += (size_t)N * 4;    // alpha_dst  [N,H]
  unsigned* mkey  = (unsigned*)(ws + o); o += (size_t)N * 4;  // segment max keys
  float*    denom = ws + o; o += (size_t)N * 4;    // softmax denominators
  float*    pbuf  = ws + o; o += (size_t)E2 * 4;   // exp(e - m) per edge/head
  (void)ws_size; (void)n_in; (void)out_size;

  const int n64   = N * 64;
  const int waves = N / 16;                        // N % 16 == 0
  const int gemmB = (waves + 7) / 8;               // 8 waves (256 thr) per block

  auto layer = [&](const float* in_feat, const float* W, const float* a_s,
                   const float* a_d, const float* b, int H, int hshift,
                   int C, int cshift, float* out_feat) {
    int nh = N * H;
    int eh = E2 * H;
    int et = E2 * 16;
    init_layer_k<<<(n64 + 255) / 256, 256, 0, stream>>>(acc, mkey, denom, n64, nh);
    gemm64_wmma_k<<<gemmB, 256, 0, stream>>>(in_feat, W, xt, N);
    attn_coeff_k<<<(nh + 255) / 256, 256, 0, stream>>>(xt, a_s, a_d, als, ald, N, H, hshift, C);
    edge_max_k<<<(eh + 255) / 256, 256, 0, stream>>>(esrc, edst, als, ald, mkey, E, E2, H, hshift);
    edge_sum_k<<<(eh + 255) / 256, 256, 0, stream>>>(esrc, edst, als, ald, mkey, denom, pbuf, E, E2, H, hshift);
    edge_agg_k<<<(et + 255) / 256, 256, 0, stream>>>(esrc, edst, xt, pbuf, denom, acc, E, E2, hshift, cshift);
    bias_act_k<<<(n64 + 255) / 256, 256, 0, stream>>>(acc, b, out_feat, n64);
  };

  // layer 0: 64 -> 4 heads x 16, concat
  layer(x,    W0, as0, ad0, b0, /*H=*/4, /*hshift=*/2, /*C=*/16, /*cshift=*/4, hbuf);
  // layer 1: 64 -> 4 heads x 16, concat
  layer(hbuf, W1, as1, ad1, b1, 4, 2, 16, 4, hbuf);
  // layer 2: 64 -> 1 head x 64, mean over heads == identity
  layer(hbuf, W2, as2, ad2, b2, /*H=*/1, /*hshift=*/0, /*C=*/64, /*cshift=*/6, hbuf);

  // final mean over nodes -> d_out[64]
  float* out = (float*)d_out;
  zero_out_k<<<1, 64, 0, stream>>>(out);
  mean_k<<<512, 256, 0, stream>>>(hbuf, out, N, 1.0f / (float)N);
}